// OneClassLoss_61942018342921
// MI455X (gfx1250) — compile-verified
//
#include <hip/hip_runtime.h>
#include <hip/hip_bf16.h>
#include <math.h>

#define BSZ 32
#define NF 64                   // 2*BS frames
#define HW 128
#define FRAME_ELEMS (HW * HW)   // 16384
#define NP 97                   // rows in logits (96 + duplicated entry)
#define NSUB 61                 // 2*BS - 3

typedef __attribute__((ext_vector_type(2))) float v2f;
typedef __attribute__((ext_vector_type(8))) float v8f;

// ---------------------------------------------------------------------------
// Compile-time port of losstemp(): rows / labels / subs tables.
// Entries are built at indices 1..96 in insertion order; the original code
// seeds with entry #10 (0-based) and then loops over all entries again, so
// that entry appears twice: we copy index 11 (orig #10) to index 0.
// ---------------------------------------------------------------------------
struct Tables {
    int rows[NP];
    int labels[NP];
    int subs[NP][NSUB];
};

constexpr int add_entry(Tables& t, int cnt, int i, int tgt,
                        int a0, int a1, int b0, int b1, int c0, int c1) {
    int n = 0;
    for (int v = a0; v < a1; ++v) t.subs[cnt][n++] = v;
    for (int v = b0; v < b1; ++v) t.subs[cnt][n++] = v;
    for (int v = c0; v < c1; ++v) t.subs[cnt][n++] = v;
    t.rows[cnt] = i;
    for (int j = 0; j < NSUB; ++j)
        if (t.subs[cnt][j] == tgt) { t.labels[cnt] = j; break; }
    return cnt + 1;
}

constexpr Tables make_tables() {
    Tables t{};
    const int bs = BSZ;
    int cnt = 1;
    for (int i = 0; i < 2 * bs; ++i) {
        if (i < bs) {
            if ((i & 1) == 0) {
                cnt = add_entry(t, cnt, i, i + 1,      0, i,     i + 1, bs + i,     bs + i + 2, 2 * bs);
                cnt = add_entry(t, cnt, i, bs + i,     0, i,     i + 2, bs + i + 1, bs + i + 2, 2 * bs);
                cnt = add_entry(t, cnt, i, bs + i + 1, 0, i,     i + 2, bs + i,     bs + i + 1, 2 * bs);
            } else {
                cnt = add_entry(t, cnt, i, bs + i - 1, 0, i - 1, i + 1, bs + i,     bs + i + 1, 2 * bs);
                cnt = add_entry(t, cnt, i, bs + i,     0, i - 1, i + 1, bs + i - 1, bs + i,     2 * bs);
            }
        } else if ((i & 1) == 0) {
            cnt = add_entry(t, cnt, i, i + 1, 0, i - bs, i - bs + 2, i, i + 1, 2 * bs);
        }
    }
    t.rows[0]   = t.rows[11];
    t.labels[0] = t.labels[11];
    for (int j = 0; j < NSUB; ++j) t.subs[0][j] = t.subs[11][j];
    return t;
}

__constant__ Tables g_tab = make_tables();

// ---------------------------------------------------------------------------
// Kernel 1: G = X * X^T via V_WMMA_F32_16X16X4_F32 (fp32 exactness required
// by the downstream softmax). 16 blocks = 4x4 tiles of 16x16; 8 waves per
// block split K=16384, partial tiles reduced through LDS.
// A-frag (16x4 f32): lane L<16 holds A[M=L][K=0..1], lane L+16 holds K=2..3.
// B-frag (4x16 f32): mirrored with N across lanes; since B = X^T the per-lane
// load pattern is identical to A's.
// ---------------------------------------------------------------------------
__device__ __forceinline__ const float* frame_ptr(const float* x1, const float* x2, int f) {
    return (f < BSZ) ? (x1 + (size_t)f * FRAME_ELEMS)
                     : (x2 + (size_t)(f - BSZ) * FRAME_ELEMS);
}

__global__ __launch_bounds__(256) void gram_wmma_kernel(const float* __restrict__ x1,
                                                        const float* __restrict__ x2,
                                                        float* __restrict__ G) {
    __shared__ float red[256 * 8];
    const int tid   = threadIdx.x;
    const int lane  = tid & 31;
    const int wv    = tid >> 5;
    const int rbase = (blockIdx.x >> 2) * 16;
    const int cbase = (blockIdx.x & 3) * 16;
    const int m     = lane & 15;
    const int koff  = (lane >> 4) << 1;   // 0 or 2

    const float* pa = frame_ptr(x1, x2, rbase + m) + koff;
    const float* pb = frame_ptr(x1, x2, cbase + m) + koff;

    v8f acc = {0.f, 0.f, 0.f, 0.f, 0.f, 0.f, 0.f, 0.f};
    const int k0 = wv * (FRAME_ELEMS / 8);
    const int k1 = k0 + (FRAME_ELEMS / 8);
#pragma unroll 8
    for (int k = k0; k < k1; k += 4) {
        v2f a = *(const v2f*)(pa + k);
        v2f b = *(const v2f*)(pb + k);
        acc = __builtin_amdgcn_wmma_f32_16x16x4_f32(
            /*neg_a=*/false, a, /*neg_b=*/false, b,
            /*c_mod=*/(short)0, acc, /*reuse_a=*/false, /*reuse_b=*/false);
    }
#pragma unroll
    for (int r = 0; r < 8; ++r) red[tid * 8 + r] = acc[r];
    __syncthreads();
    if (wv == 0) {
#pragma unroll
        for (int r = 0; r < 8; ++r) {
            float s = 0.f;
            for (int w = 0; w < 8; ++w) s += red[(w * 32 + lane) * 8 + r];
            const int mm = r + ((lane >> 4) << 3);  // C/D layout: VGPR r, lane<16 -> M=r, lane>=16 -> M=r+8
            const int nn = lane & 15;
            G[(rbase + mm) * NF + (cbase + nn)] = s;
        }
    }
}

// ---------------------------------------------------------------------------
// Shared-memory 128-point radix-2 DIT FFT (forward, e^{-i...} like jnp.fft).
// 128 threads; input in natural order at re[t]/im[t]; output natural order.
// ---------------------------------------------------------------------------
__device__ __forceinline__ void fft128(float* __restrict__ re, float* __restrict__ im, int t) {
    const int rv = (int)(__brev((unsigned)t) >> 25);   // 7-bit bit reversal
    const float vr = re[rv], vi = im[rv];
    __syncthreads();
    re[t] = vr; im[t] = vi;
    __syncthreads();
#pragma unroll
    for (int s = 0; s < 7; ++s) {
        const int mh = 1 << s;
        if (t < 64) {
            const int grp = t >> s;
            const int pos = t & (mh - 1);
            const int i0  = (grp << (s + 1)) + pos;
            const int i1  = i0 + mh;
            const float ang = -3.14159265358979323846f * (float)pos / (float)mh;
            const float wr = cosf(ang);
            const float wi = sinf(ang);
            const float xr = re[i1], xi = im[i1];
            const float tr = wr * xr - wi * xi;
            const float ti = wr * xi + wi * xr;
            const float ur = re[i0], ui = im[i0];
            re[i1] = ur - tr; im[i1] = ui - ti;
            re[i0] = ur + tr; im[i0] = ui + ti;
        }
        __syncthreads();
    }
}

// Row FFTs: one block per (frame,row); transposed store -> inter[frame][col][row]
__global__ __launch_bounds__(128) void fft_rows_kernel(const float* __restrict__ x1,
                                                       const float* __restrict__ x2,
                                                       float* __restrict__ ire,
                                                       float* __restrict__ iim) {
    __shared__ float re[HW], im[HW];
    const int t     = threadIdx.x;
    const int frame = blockIdx.x >> 7;
    const int row   = blockIdx.x & 127;
    const float* src = frame_ptr(x1, x2, frame);
    re[t] = src[row * HW + t];
    im[t] = 0.f;
    __syncthreads();
    fft128(re, im, t);
    ire[((size_t)frame * HW + t) * HW + row] = re[t];
    iim[((size_t)frame * HW + t) * HW + row] = im[t];
}

// Column FFTs + |.|^2 accumulation over all 64 frames (no atomics needed).
// The psd bin permutation is irrelevant: downstream reductions are
// permutation-invariant (mean(log) and log(mean) over all 16384 bins).
__global__ __launch_bounds__(128) void fft_cols_kernel(const float* __restrict__ ire,
                                                       const float* __restrict__ iim,
                                                       float* __restrict__ psd) {
    __shared__ float re[HW], im[HW];
    const int t = threadIdx.x;
    const int c = blockIdx.x;
    float acc = 0.f;
    for (int f = 0; f < NF; ++f) {
        __syncthreads();
        re[t] = ire[((size_t)f * HW + c) * HW + t];
        im[t] = iim[((size_t)f * HW + c) * HW + t];
        __syncthreads();
        fft128(re, im, t);
        acc += re[t] * re[t] + im[t] * im[t];
    }
    psd[c * HW + t] = acc * (1.0f / NF);   // mean over frames
}

// ---------------------------------------------------------------------------
// Final: PSD reductions + cross-entropy from the Gram matrix.
// logits'[p][j] = G[s][s] - 2 G[r][s]  (G[r][r] cancels in log_softmax)
// ---------------------------------------------------------------------------
__global__ __launch_bounds__(128) void final_kernel(const float* __restrict__ G,
                                                    const float* __restrict__ psd,
                                                    float* __restrict__ out) {
    __shared__ float sv[128], sl[128], sc[128];
    const int t = threadIdx.x;

    float accv = 0.f, accl = 0.f;
    for (int i = t; i < FRAME_ELEMS; i += 128) {
        const float v = psd[i];
        accv += v;
        accl += logf(v);
    }

    float ce = 0.f;
    if (t < NP) {
        const int r   = g_tab.rows[t];
        const int lbl = g_tab.labels[t];
        float mx = -3.4e38f;
        for (int j = 0; j < NSUB; ++j) {
            const int s = g_tab.subs[t][j];
            const float lg = G[s * NF + s] - 2.f * G[r * NF + s];
            mx = fmaxf(mx, lg);
        }
        float se = 0.f, ltgt = 0.f;
        for (int j = 0; j < NSUB; ++j) {
            const int s = g_tab.subs[t][j];
            const float lg = G[s * NF + s] - 2.f * G[r * NF + s];
            se += expf(lg - mx);
            if (j == lbl) ltgt = lg;
        }
        ce = -(ltgt - (mx + logf(se)));
    }

    sv[t] = accv; sl[t] = accl; sc[t] = ce;
    __syncthreads();
    for (int off = 64; off > 0; off >>= 1) {
        if (t < off) { sv[t] += sv[t + off]; sl[t] += sl[t + off]; sc[t] += sc[t + off]; }
        __syncthreads();
    }
    if (t == 0) {
        const float l1       = sc[0] / (float)NP;
        const float mean_log = sl[0] / (float)FRAME_ELEMS;
        const float log_mean = logf(sv[0] / (float)FRAME_ELEMS);
        const float l2       = -0.5f * (mean_log - log_mean);   // -REG * psd
        out[0] = l1 + l2;
    }
}

// ---------------------------------------------------------------------------
// Launch. Workspace layout (floats): G[4096] | psd[16384] | ire[64*16384] |
// iim[64*16384]  (~8.1 MB total).
// ---------------------------------------------------------------------------
extern "C" void kernel_launch(void* const* d_in, const int* in_sizes, int n_in,
                              void* d_out, int out_size, void* d_ws, size_t ws_size,
                              hipStream_t stream) {
    const float* x1 = (const float*)d_in[0];
    const float* x2 = (const float*)d_in[1];
    float* ws  = (float*)d_ws;
    float* G   = ws;                                    // 4096
    float* psd = G + NF * NF;                           // 16384
    float* ire = psd + FRAME_ELEMS;                     // 64*16384
    float* iim = ire + (size_t)NF * FRAME_ELEMS;        // 64*16384

    gram_wmma_kernel<<<16, 256, 0, stream>>>(x1, x2, G);
    fft_rows_kernel<<<NF * HW, HW, 0, stream>>>(x1, x2, ire, iim);
    fft_cols_kernel<<<HW, HW, 0, stream>>>(ire, iim, psd);
    final_kernel<<<1, 128, 0, stream>>>(G, psd, (float*)d_out);
}